// CrystalGraphConv_44427141710551
// MI455X (gfx1250) — compile-verified
//
#include <hip/hip_runtime.h>
#include <hip/hip_bf16.h>

// CrystalGraphConv for MI455X (gfx1250), fp32 WMMA path (V_WMMA_F32_16X16X4_F32).
// Edge kernel: 32 edges/block (2 M-tiles per wave) so each B fragment feeds 2 WMMAs.

#define HID     128
#define EDIM    64
#define KIN     320            // 2*HID + EDIM
#define NN      10000
#define NE      320000
#define MT      32             // edges per block (2 x 16-row M-tiles)
#define BN_EPS  1e-5f

typedef float v2f __attribute__((ext_vector_type(2)));
typedef float v8f __attribute__((ext_vector_type(8)));

#define WMMA_F32X4(A, B, C) \
  __builtin_amdgcn_wmma_f32_16x16x4_f32(false, (A), false, (B), (short)0, (C), false, false)

#define V8F_ZERO {0.f,0.f,0.f,0.f,0.f,0.f,0.f,0.f}

// -------------------------------------------------------------------------
// Kernel 1: per 32-edge tile: gather, edge MLP (WMMA), sigmoid gate, scatter
// -------------------------------------------------------------------------
__global__ __launch_bounds__(256) void edge_kernel(
    const float* __restrict__ nodef,   // [NN, HID]
    const float* __restrict__ edgef,   // [NE, EDIM]
    const float* __restrict__ We1,     // [KIN, HID]
    const float* __restrict__ be1,     // [HID]
    const float* __restrict__ We2,     // [HID, HID]
    const float* __restrict__ be2,     // [HID]
    const int*   __restrict__ eidx,    // [2, NE]
    float*       __restrict__ agg)     // [NN, HID]
{
  __shared__ float Ein[MT][KIN + 4];   // 324-float rows -> 16B-aligned rows
  __shared__ float Hs[MT][HID + 4];    // 132-float rows
  __shared__ int   sidx[MT], didx[MT];

  const int t       = threadIdx.x;
  const int lane    = t & 31;
  const int wid     = t >> 5;          // 0..7: this wave's 16-col stripe
  const int half    = lane >> 4;
  const int l16     = lane & 15;
  const int twoHalf = half * 2;
  const int col     = wid * 16 + l16;  // output column 0..127

  const int e0 = blockIdx.x * MT;

  if (t < MT) {
    sidx[t] = eidx[e0 + t];
    didx[t] = eidx[NE + e0 + t];
  }
  __syncthreads();

  // ---- stage edge_input tile [32 x 320] = [src | dst | edge] ----
  #pragma unroll
  for (int i = t; i < 1024; i += 256) {            // src: 32 rows x 32 float4
    const int r = i >> 5, q = i & 31;
    const float4 v = *(reinterpret_cast<const float4*>(nodef + (size_t)sidx[r] * HID) + q);
    float* d = &Ein[r][q * 4];
    d[0] = v.x; d[1] = v.y; d[2] = v.z; d[3] = v.w;
  }
  #pragma unroll
  for (int i = t; i < 1024; i += 256) {            // dst
    const int r = i >> 5, q = i & 31;
    const float4 v = *(reinterpret_cast<const float4*>(nodef + (size_t)didx[r] * HID) + q);
    float* d = &Ein[r][HID + q * 4];
    d[0] = v.x; d[1] = v.y; d[2] = v.z; d[3] = v.w;
  }
  #pragma unroll
  for (int i = t; i < 512; i += 256) {             // edge feats: 32 rows x 16 float4
    const int r = i >> 4, q = i & 15;
    const float4 v = *(reinterpret_cast<const float4*>(edgef + (size_t)(e0 + r) * EDIM) + q);
    float* d = &Ein[r][2 * HID + q * 4];
    d[0] = v.x; d[1] = v.y; d[2] = v.z; d[3] = v.w;
  }
  __syncthreads();

  // ---- layer 1: h = relu(Ein @ We1 + be1), K = 320 ----
  // 2 M-tiles x 2 K-chains = 4 independent accumulator chains; B reused 2x.
  v8f a00 = V8F_ZERO, a01 = V8F_ZERO;   // M-tile 0 (rows 0..15)
  v8f a10 = V8F_ZERO, a11 = V8F_ZERO;   // M-tile 1 (rows 16..31)
  #pragma unroll 2
  for (int k = 0; k < KIN; k += 8) {
    v2f b0, b1;
    {
      const float* wp0 = We1 + (size_t)(k + twoHalf) * HID + col;
      const float* wp1 = We1 + (size_t)(k + 4 + twoHalf) * HID + col;
      b0.x = wp0[0]; b0.y = wp0[HID];
      b1.x = wp1[0]; b1.y = wp1[HID];
    }
    v2f p00, p01, p10, p11;
    p00.x = Ein[l16][k + twoHalf];           p00.y = Ein[l16][k + twoHalf + 1];
    p01.x = Ein[l16][k + 4 + twoHalf];       p01.y = Ein[l16][k + 4 + twoHalf + 1];
    p10.x = Ein[16 + l16][k + twoHalf];      p10.y = Ein[16 + l16][k + twoHalf + 1];
    p11.x = Ein[16 + l16][k + 4 + twoHalf];  p11.y = Ein[16 + l16][k + 4 + twoHalf + 1];
    a00 = WMMA_F32X4(p00, b0, a00);
    a10 = WMMA_F32X4(p10, b0, a10);
    a01 = WMMA_F32X4(p01, b1, a01);
    a11 = WMMA_F32X4(p11, b1, a11);
  }
  {
    const v8f h0 = a00 + a01;
    const v8f h1 = a10 + a11;
    const float bb = be1[col];
    #pragma unroll
    for (int v = 0; v < 8; ++v) {
      // C layout: VGPR v -> row M = half*8 + v, column = col
      Hs[half * 8 + v][col]      = fmaxf(h0[v] + bb, 0.0f);
      Hs[16 + half * 8 + v][col] = fmaxf(h1[v] + bb, 0.0f);
    }
  }
  __syncthreads();

  // ---- layer 2: ew = sigmoid(h @ We2 + be2), K = 128 ----
  v8f c00 = V8F_ZERO, c01 = V8F_ZERO;
  v8f c10 = V8F_ZERO, c11 = V8F_ZERO;
  #pragma unroll 2
  for (int k = 0; k < HID; k += 8) {
    v2f b0, b1;
    {
      const float* wp0 = We2 + (size_t)(k + twoHalf) * HID + col;
      const float* wp1 = We2 + (size_t)(k + 4 + twoHalf) * HID + col;
      b0.x = wp0[0]; b0.y = wp0[HID];
      b1.x = wp1[0]; b1.y = wp1[HID];
    }
    v2f p00, p01, p10, p11;
    p00.x = Hs[l16][k + twoHalf];           p00.y = Hs[l16][k + twoHalf + 1];
    p01.x = Hs[l16][k + 4 + twoHalf];       p01.y = Hs[l16][k + 4 + twoHalf + 1];
    p10.x = Hs[16 + l16][k + twoHalf];      p10.y = Hs[16 + l16][k + twoHalf + 1];
    p11.x = Hs[16 + l16][k + 4 + twoHalf];  p11.y = Hs[16 + l16][k + 4 + twoHalf + 1];
    c00 = WMMA_F32X4(p00, b0, c00);
    c10 = WMMA_F32X4(p10, b0, c10);
    c01 = WMMA_F32X4(p01, b1, c01);
    c11 = WMMA_F32X4(p11, b1, c11);
  }
  {
    const v8f w0 = c00 + c01;
    const v8f w1 = c10 + c11;
    const float bb = be2[col];
    #pragma unroll
    for (int v = 0; v < 8; ++v) {
      {
        const int m = half * 8 + v;                    // edge row in tile (M-tile 0)
        const float x  = w0[v] + bb;
        const float ew = 1.0f / (1.0f + __expf(-x));   // sigmoid
        const float msg = Ein[m][col] * ew;            // gate src_feat (cols 0..127)
        __hip_atomic_fetch_add(&agg[(size_t)didx[m] * HID + col], msg,
                               __ATOMIC_RELAXED, __HIP_MEMORY_SCOPE_AGENT);
      }
      {
        const int m = 16 + half * 8 + v;               // M-tile 1
        const float x  = w1[v] + bb;
        const float ew = 1.0f / (1.0f + __expf(-x));
        const float msg = Ein[m][col] * ew;
        __hip_atomic_fetch_add(&agg[(size_t)didx[m] * HID + col], msg,
                               __ATOMIC_RELAXED, __HIP_MEMORY_SCOPE_AGENT);
      }
    }
  }
}

// -------------------------------------------------------------------------
// Kernel 2: per 16-node tile: node MLP (WMMA) -> updated (pre-BN) + BN stats
// -------------------------------------------------------------------------
__global__ __launch_bounds__(256) void node_kernel(
    const float* __restrict__ nodef,   // [NN, HID]
    const float* __restrict__ agg,     // [NN, HID]
    const float* __restrict__ Wn1,     // [2*HID, HID]
    const float* __restrict__ bn1,     // [HID]
    const float* __restrict__ Wn2,     // [HID, HID]
    const float* __restrict__ bn2,     // [HID]
    float*       __restrict__ updated, // [NN, HID]  (d_out, pre-BN)
    float*       __restrict__ colsum,  // [HID]
    float*       __restrict__ colsumsq)// [HID]
{
  __shared__ float Nin[16][2 * HID + 4];  // 260-float rows
  __shared__ float Hs[16][HID + 4];
  __shared__ float bsum[HID], bsq[HID];

  const int t       = threadIdx.x;
  const int lane    = t & 31;
  const int wid     = t >> 5;
  const int half    = lane >> 4;
  const int l16     = lane & 15;
  const int twoHalf = half * 2;
  const int col     = wid * 16 + l16;

  const int n0 = blockIdx.x * 16;

  if (t < HID) { bsum[t] = 0.0f; bsq[t] = 0.0f; }

  // ---- stage node_input tile [16 x 256] = [node | agg] ----
  #pragma unroll
  for (int i = t; i < 1024; i += 256) {   // 16 rows x 64 float4
    const int r = i >> 6, q = i & 63;
    const float4 v = (q < 32)
      ? *(reinterpret_cast<const float4*>(nodef + (size_t)(n0 + r) * HID) + q)
      : *(reinterpret_cast<const float4*>(agg   + (size_t)(n0 + r) * HID) + (q - 32));
    float* d = &Nin[r][q * 4];
    d[0] = v.x; d[1] = v.y; d[2] = v.z; d[3] = v.w;
  }
  __syncthreads();

  // ---- layer 1: relu(Nin @ Wn1 + bn1), K = 256 ----
  v8f acc0 = V8F_ZERO, acc1 = V8F_ZERO;
  #pragma unroll 4
  for (int k = 0; k < 2 * HID; k += 8) {
    {
      v2f a, b;
      a.x = Nin[l16][k + twoHalf];
      a.y = Nin[l16][k + twoHalf + 1];
      const float* wp = Wn1 + (size_t)(k + twoHalf) * HID + col;
      b.x = wp[0]; b.y = wp[HID];
      acc0 = WMMA_F32X4(a, b, acc0);
    }
    {
      v2f a, b;
      a.x = Nin[l16][k + 4 + twoHalf];
      a.y = Nin[l16][k + 4 + twoHalf + 1];
      const float* wp = Wn1 + (size_t)(k + 4 + twoHalf) * HID + col;
      b.x = wp[0]; b.y = wp[HID];
      acc1 = WMMA_F32X4(a, b, acc1);
    }
  }
  {
    const v8f h = acc0 + acc1;
    const float b1 = bn1[col];
    #pragma unroll
    for (int v = 0; v < 8; ++v)
      Hs[half * 8 + v][col] = fmaxf(h[v] + b1, 0.0f);
  }
  __syncthreads();

  // ---- layer 2: h @ Wn2 + bn2, K = 128 ----
  v8f c0 = V8F_ZERO, c1 = V8F_ZERO;
  #pragma unroll 4
  for (int k = 0; k < HID; k += 8) {
    {
      v2f a, b;
      a.x = Hs[l16][k + twoHalf];
      a.y = Hs[l16][k + twoHalf + 1];
      const float* wp = Wn2 + (size_t)(k + twoHalf) * HID + col;
      b.x = wp[0]; b.y = wp[HID];
      c0 = WMMA_F32X4(a, b, c0);
    }
    {
      v2f a, b;
      a.x = Hs[l16][k + 4 + twoHalf];
      a.y = Hs[l16][k + 4 + twoHalf + 1];
      const float* wp = Wn2 + (size_t)(k + 4 + twoHalf) * HID + col;
      b.x = wp[0]; b.y = wp[HID];
      c1 = WMMA_F32X4(a, b, c1);
    }
  }
  {
    const v8f u = c0 + c1;
    const float b2 = bn2[col];
    float s = 0.0f, sq = 0.0f;
    #pragma unroll
    for (int v = 0; v < 8; ++v) {
      const int m = half * 8 + v;
      const float x = u[v] + b2;
      updated[(size_t)(n0 + m) * HID + col] = x;
      s  += x;
      sq += x * x;
    }
    __hip_atomic_fetch_add(&bsum[col], s,  __ATOMIC_RELAXED, __HIP_MEMORY_SCOPE_WORKGROUP);
    __hip_atomic_fetch_add(&bsq[col],  sq, __ATOMIC_RELAXED, __HIP_MEMORY_SCOPE_WORKGROUP);
  }
  __syncthreads();

  if (t < HID) {
    __hip_atomic_fetch_add(&colsum[t],   bsum[t], __ATOMIC_RELAXED, __HIP_MEMORY_SCOPE_AGENT);
    __hip_atomic_fetch_add(&colsumsq[t], bsq[t],  __ATOMIC_RELAXED, __HIP_MEMORY_SCOPE_AGENT);
  }
}

// -------------------------------------------------------------------------
// Kernel 3: BatchNorm (training-mode batch stats) + residual, in place
// -------------------------------------------------------------------------
__global__ __launch_bounds__(256) void finalize_kernel(
    const float* __restrict__ nodef,
    const float* __restrict__ gamma,
    const float* __restrict__ beta,
    const float* __restrict__ colsum,
    const float* __restrict__ colsumsq,
    float*       __restrict__ out)
{
  const int i = blockIdx.x * 256 + threadIdx.x;
  if (i >= NN * HID) return;
  const int c = i & (HID - 1);
  const float inv_n = 1.0f / (float)NN;
  const float mean  = colsum[c] * inv_n;
  const float var   = colsumsq[c] * inv_n - mean * mean;
  const float u     = out[i];
  out[i] = nodef[i] + (u - mean) * rsqrtf(var + BN_EPS) * gamma[c] + beta[c];
}

// -------------------------------------------------------------------------
extern "C" void kernel_launch(void* const* d_in, const int* in_sizes, int n_in,
                              void* d_out, int out_size, void* d_ws, size_t ws_size,
                              hipStream_t stream) {
  const float* nodef = (const float*)d_in[0];
  const float* edgef = (const float*)d_in[1];
  const float* We1   = (const float*)d_in[2];
  const float* be1   = (const float*)d_in[3];
  const float* We2   = (const float*)d_in[4];
  const float* be2   = (const float*)d_in[5];
  const float* Wn1   = (const float*)d_in[6];
  const float* bn1   = (const float*)d_in[7];
  const float* Wn2   = (const float*)d_in[8];
  const float* bn2   = (const float*)d_in[9];
  const float* gamma = (const float*)d_in[10];
  const float* beta  = (const float*)d_in[11];
  const int*   eidx  = (const int*)d_in[12];

  float* out      = (float*)d_out;                 // [NN, HID]
  float* agg      = (float*)d_ws;                  // [NN, HID]
  float* colsum   = agg + (size_t)NN * HID;        // [HID]
  float* colsumsq = colsum + HID;                  // [HID]

  // zero aggregation buffer + stats (graph-capture-safe async memset)
  hipMemsetAsync(d_ws, 0, ((size_t)NN * HID + 2 * HID) * sizeof(float), stream);

  edge_kernel<<<NE / MT, 256, 0, stream>>>(nodef, edgef, We1, be1, We2, be2, eidx, agg);
  node_kernel<<<NN / 16, 256, 0, stream>>>(nodef, agg, Wn1, bn1, Wn2, bn2, out, colsum, colsumsq);
  finalize_kernel<<<(NN * HID + 255) / 256, 256, 0, stream>>>(nodef, gamma, beta, colsum, colsumsq, out);
}